// ConnectedFilterLayer_80384607912362
// MI455X (gfx1250) — compile-verified
//
#include <hip/hip_runtime.h>

typedef __attribute__((ext_vector_type(2))) float v2f;
typedef __attribute__((ext_vector_type(8))) float v8f;

#define BETA_F 1.0f
#define DOUBLING_ITERS 21

// ---------------------------------------------------------------------------
// Kernel 1: per-wave 16-node tile of the N x 8 matvec via V_WMMA_F32_16X16X4_F32
// (K=8 split into two K=4 WMMA accumulations), then sigmoid + tree-contrib.
//
// A-fragment layout (ISA 7.12.2, 32-bit A 16x4): lane m=lane&15 holds row m;
// VGPR0/VGPR1 hold K = 2*half / 2*half+1 (half = lane>>4).
// B-fragment: weight replicated across all 16 columns, so D[m][n] = dot(A_m, w)
// for every column n -> epilogue does not depend on the B column striping.
// C/D layout: VGPR r = row r (lanes 0-15) and row r+8 (lanes 16-31).
// ---------------------------------------------------------------------------
__global__ void cf_contrib_wmma(const float* __restrict__ attrs,
                                const float* __restrict__ w,
                                const float* __restrict__ bias,
                                const float* __restrict__ levels,
                                const int*   __restrict__ parent,
                                float* __restrict__ s_out,
                                int*   __restrict__ p_out,
                                int N, int numTiles)
{
    const int lane = threadIdx.x & 31;     // wave32
    const int half = lane >> 4;            // 0: lanes 0-15, 1: lanes 16-31
    const int m    = lane & 15;            // A-matrix row within tile
    const int wavesPerBlock = blockDim.x >> 5;
    const int waveId     = blockIdx.x * wavesPerBlock + (threadIdx.x >> 5);
    const int waveStride = gridDim.x * wavesPerBlock;

    // B fragments (uniform per lane-half): K = 2*half, 2*half+1 for each chunk.
    v2f b0, b1;
    b0.x = w[half * 2 + 0]; b0.y = w[half * 2 + 1];
    b1.x = w[half * 2 + 4]; b1.y = w[half * 2 + 5];
    const float bia = bias[0];

    for (int tile = waveId; tile < numTiles; tile += waveStride) {
        const int base = tile << 4;
        int row = base + m;
        if (row >= N) row = N - 1;          // clamp (keeps EXEC all-1 for WMMA)
        const float* ap = attrs + (size_t)row * 8;

        v2f a0, a1;
        a0.x = ap[half * 2 + 0]; a0.y = ap[half * 2 + 1];
        a1.x = ap[half * 2 + 4]; a1.y = ap[half * 2 + 5];

        v8f c = {0.f, 0.f, 0.f, 0.f, 0.f, 0.f, 0.f, 0.f};
        c = __builtin_amdgcn_wmma_f32_16x16x4_f32(false, a0, false, b0,
                                                  (short)0, c, false, false);
        c = __builtin_amdgcn_wmma_f32_16x16x4_f32(false, a1, false, b1,
                                                  (short)0, c, false, false);

        // lanes 0-15 hold rows base+0..7 in c[0..7]; lanes 16-31 rows base+8..15.
        if ((lane & 15) < 8) {
            const int r = lane & 7;
            float dotv = c[0];
            if (r == 1) dotv = c[1];
            if (r == 2) dotv = c[2];
            if (r == 3) dotv = c[3];
            if (r == 4) dotv = c[4];
            if (r == 5) dotv = c[5];
            if (r == 6) dotv = c[6];
            if (r == 7) dotv = c[7];

            const int node = base + half * 8 + r;
            if (node < N) {
                const float x  = BETA_F * (dotv + bia);
                // sigmoid via single v_exp_f32 + v_rcp_f32 (avoid IEEE div sequence)
                const float sg = __builtin_amdgcn_rcpf(1.0f + __expf(-x));
                const int   par = parent[node];
                float contrib = sg * (levels[node] - levels[par]);
                int   pv = par;
                if (node == 0) { contrib = levels[0]; pv = N; }  // root + sentinel parent
                s_out[node] = contrib;
                p_out[node] = pv;
            }
        }
    }

    if (blockIdx.x == 0 && threadIdx.x == 0) {
        s_out[N] = 0.0f;   // sentinel node
        p_out[N] = N;      // self-parent
    }
}

// ---------------------------------------------------------------------------
// Kernel 2: one pointer-doubling step (functional update -> ping-pong buffers).
// Working set (~16 MB) fits the 192 MB L2, so the random gathers hit L2.
// ---------------------------------------------------------------------------
__global__ void cf_double(const float* __restrict__ s_in,
                          const int*   __restrict__ p_in,
                          float* __restrict__ s_out,
                          int*   __restrict__ p_out,
                          int n)
{
    const int i = blockIdx.x * blockDim.x + threadIdx.x;
    if (i < n) {
        const int pi = p_in[i];
        s_out[i] = s_in[i] + s_in[pi];
        p_out[i] = p_in[pi];
    }
}

// ---------------------------------------------------------------------------
// Kernel 3: out[j] = s[pixel_node[j]]
// ---------------------------------------------------------------------------
__global__ void cf_gather(const float* __restrict__ s,
                          const int*   __restrict__ pix,
                          float* __restrict__ out,
                          int M)
{
    const int i = blockIdx.x * blockDim.x + threadIdx.x;
    if (i < M) out[i] = s[pix[i]];
}

extern "C" void kernel_launch(void* const* d_in, const int* in_sizes, int n_in,
                              void* d_out, int out_size, void* d_ws, size_t ws_size,
                              hipStream_t stream)
{
    const float* attrs  = (const float*)d_in[0];   // N x 8
    const float* weight = (const float*)d_in[1];   // 8
    const float* bias   = (const float*)d_in[2];   // 1
    const float* levels = (const float*)d_in[3];   // N
    const int*   parent = (const int*)  d_in[4];   // N
    const int*   pixel  = (const int*)  d_in[5];   // H*W
    float*       out    = (float*)d_out;

    const int N = in_sizes[3];
    const int M = out_size;
    const size_t n1 = (size_t)N + 1;

    // Workspace: 4 ping-pong buffers (s x2, p x2), 256B-aligned segments.
    const size_t seg = ((n1 * 4 + 255) / 256) * 256;
    char*  base = (char*)d_ws;
    float* sA = (float*)(base + 0 * seg);
    float* sB = (float*)(base + 1 * seg);
    int*   pA = (int*)  (base + 2 * seg);
    int*   pB = (int*)  (base + 3 * seg);
    (void)ws_size; (void)n_in;

    // Phase 1: WMMA matvec + sigmoid + contrib
    const int numTiles = (N + 15) / 16;
    const int wavesPerBlock = 8;               // 256 threads = 8 wave32s
    const int blocks1 = (numTiles + wavesPerBlock - 1) / wavesPerBlock;
    cf_contrib_wmma<<<blocks1, 256, 0, stream>>>(attrs, weight, bias, levels,
                                                 parent, sA, pA, N, numTiles);

    // Phase 2: 21 pointer-doubling steps, ping-ponging buffers
    float* s_in = sA; float* s_out_ = sB;
    int*   p_in = pA; int*   p_out_ = pB;
    const int n = (int)n1;
    const int blocks2 = (n + 255) / 256;
    for (int it = 0; it < DOUBLING_ITERS; ++it) {
        cf_double<<<blocks2, 256, 0, stream>>>(s_in, p_in, s_out_, p_out_, n);
        float* ts = s_in; s_in = s_out_; s_out_ = ts;
        int*   tp = p_in; p_in = p_out_; p_out_ = tp;
    }

    // Phase 3: pixel gather
    const int blocks3 = (M + 255) / 256;
    cf_gather<<<blocks3, 256, 0, stream>>>(s_in, pixel, out, M);
}